// WaveRNN_60035052864136
// MI455X (gfx1250) — compile-verified
//
#include <hip/hip_runtime.h>

// MI455X / gfx1250, wave32.
// 2-D wave equation: one workgroup per batch, whole 160x160 field resident in LDS,
// Laplacian = A*H + H*A with A = tridiag(1,-2,1) via V_WMMA_F32_16X16X4_F32.
// All time-invariant state (tile coords, coeff, source predicate, A16 operand)
// hoisted into registers; time loop is pure LDS + WMMA + VALU, one barrier/step.

typedef __attribute__((ext_vector_type(2))) float v2f;
typedef __attribute__((ext_vector_type(8))) float v8f;

#define NN     160          // grid size
#define STRIDE 164          // LDS row stride (floats): conflict-free for row & column access
#define TSTEPS 512
#define NB     4
#define NP     32
#define NTILES 100          // 10x10 tiles of 16x16
#define NWAVES 32           // 1024 threads / wave32
#define MAXK   4            // ceil(NTILES / NWAVES)

__global__ __launch_bounds__(1024, 1)
void wave_kernel(const float* __restrict__ x,        // (B,T)
                 const float* __restrict__ cmat,     // (160,160)
                 const float* __restrict__ rho,      // (160,160)
                 const int* __restrict__ src_i_p,
                 const int* __restrict__ src_j_p,
                 const int* __restrict__ probe_i,
                 const int* __restrict__ probe_j,
                 float* __restrict__ out)             // (B,T,P)
{
    __shared__ float buf[2][NN * STRIDE];            // 209920 B

    const int b    = blockIdx.x;
    const int tid  = threadIdx.x;
    const int lane = tid & 31;
    const int wave = tid >> 5;
    const int m    = lane & 15;      // column (C/B layout) or row (A layout) in tile
    const int hh   = lane >> 4;      // lane half

    // zero both LDS buffers (h0 = 0)
    for (int i = tid; i < 2 * NN * STRIDE; i += 1024)
        ((float*)buf)[i] = 0.0f;

    const int src_i = src_i_p[0];
    const int src_j = src_j_p[0];

    int pi = 0, pj = 0;
    if (wave == 0) { pi = probe_i[lane]; pj = probe_j[lane]; }

    // A16 = tridiag(1,-2,1) as four 16x4 K-chunks in WMMA A-operand layout
    // (lane -> M = lane%16; VGPR p, lane-half h -> K = 4c + p + 2h).
    // A16 is symmetric, so the same registers are the B-operand (rows of A16).
    v2f Aop[4];
#pragma unroll
    for (int cc = 0; cc < 4; ++cc) {
        int k0 = 4 * cc + 2 * hh, k1 = k0 + 1;
        Aop[cc].x = (m == k0) ? -2.0f : (((m - k0 == 1) || (k0 - m == 1)) ? 1.0f : 0.0f);
        Aop[cc].y = (m == k1) ? -2.0f : (((m - k1 == 1) || (k1 - m == 1)) ? 1.0f : 0.0f);
    }

    // ---- hoist all time-invariant per-tile state into registers ----
    int   tgi[MAXK], tgj[MAXK];
    float creg[MAXK][8];             // coeff = 0.16*c*c/rho at this thread's elements
    int   src_code = -1;             // k*8+r if this thread owns the source point
#pragma unroll
    for (int k = 0; k < MAXK; ++k) {
        const int tile = wave + NWAVES * k;
        tgi[k] = 0; tgj[k] = 0;
        if (tile < NTILES) {
            const int I = tile / 10, J = tile % 10;
            tgi[k] = I * 16; tgj[k] = J * 16;
#pragma unroll
            for (int r = 0; r < 8; ++r) {
                const int row = tgi[k] + r + 8 * hh;
                const int col = tgj[k] + m;
                const float cv = cmat[row * NN + col];
                creg[k][r] = (0.16f * cv) * cv / rho[row * NN + col];
                if (row == src_i && col == src_j) src_code = k * 8 + r;
            }
        } else {
#pragma unroll
            for (int r = 0; r < 8; ++r) creg[k][r] = 0.0f;
        }
    }

    __syncthreads();

    // one time step: read h1 (untouched), read+overwrite h2 with hn (own elements only)
    auto step = [&](const float* __restrict__ h1, float* __restrict__ h2, int t) {
        const float xv = x[b * TSTEPS + t];
#pragma unroll
        for (int k = 0; k < MAXK; ++k) {
            const int tile = wave + NWAVES * k;     // wave-uniform guard: EXEC all-1 inside
            if (tile < NTILES) {
                const int gi = tgi[k], gj = tgj[k];

                // two independent accumulator chains for matrix-pipe ILP
                v8f acc0 = {0.f, 0.f, 0.f, 0.f, 0.f, 0.f, 0.f, 0.f};
                v8f acc1 = {0.f, 0.f, 0.f, 0.f, 0.f, 0.f, 0.f, 0.f};

                // acc += A16 * H  (vertical 2nd difference, tile interior)
#pragma unroll
                for (int cc = 0; cc < 4; cc += 2) {
                    v2f Hb0, Hb1;   // B-operand: K-rows of H at column gj+m
                    Hb0.x = h1[(gi + 4 * cc + 2 * hh) * STRIDE + gj + m];
                    Hb0.y = h1[(gi + 4 * cc + 1 + 2 * hh) * STRIDE + gj + m];
                    Hb1.x = h1[(gi + 4 * (cc + 1) + 2 * hh) * STRIDE + gj + m];
                    Hb1.y = h1[(gi + 4 * (cc + 1) + 1 + 2 * hh) * STRIDE + gj + m];
                    acc0 = __builtin_amdgcn_wmma_f32_16x16x4_f32(
                        false, Aop[cc], false, Hb0, (short)0, acc0, false, false);
                    acc1 = __builtin_amdgcn_wmma_f32_16x16x4_f32(
                        false, Aop[cc + 1], false, Hb1, (short)0, acc1, false, false);
                }
                // acc += H * A16  (horizontal 2nd difference)
#pragma unroll
                for (int cc = 0; cc < 4; cc += 2) {
                    // A-operand: adjacent columns of H -> aligned 8B LDS loads
                    v2f Ha0 = *(const v2f*)&h1[(gi + m) * STRIDE + gj + 4 * cc + 2 * hh];
                    v2f Ha1 = *(const v2f*)&h1[(gi + m) * STRIDE + gj + 4 * (cc + 1) + 2 * hh];
                    acc0 = __builtin_amdgcn_wmma_f32_16x16x4_f32(
                        false, Ha0, false, Aop[cc], (short)0, acc0, false, false);
                    acc1 = __builtin_amdgcn_wmma_f32_16x16x4_f32(
                        false, Ha1, false, Aop[cc + 1], (short)0, acc1, false, false);
                }

                // rank-1 halo terms across tile boundaries (zero BC at grid edge)
                if (gi > 0 && hh == 0)        acc0[0] += h1[(gi - 1) * STRIDE + gj + m];
                if (gi + 16 < NN && hh == 1)  acc0[7] += h1[(gi + 16) * STRIDE + gj + m];
                if (gj > 0 && m == 0) {
#pragma unroll
                    for (int r = 0; r < 8; ++r)
                        acc0[r] += h1[(gi + r + 8 * hh) * STRIDE + gj - 1];
                }
                if (gj + 16 < NN && m == 15) {
#pragma unroll
                    for (int r = 0; r < 8; ++r)
                        acc0[r] += h1[(gi + r + 8 * hh) * STRIDE + gj + 16];
                }

                // hn = 2*h1 - h2 + coeff*lap (+ source), in place of h2
#pragma unroll
                for (int r = 0; r < 8; ++r) {
                    const int a = (gi + r + 8 * hh) * STRIDE + gj + m;
                    const float h1v = h1[a];
                    const float h2v = h2[a];
                    float hn = 2.0f * h1v - h2v + creg[k][r] * (acc0[r] + acc1[r]);
                    hn += (src_code == k * 8 + r) ? xv : 0.0f;   // branchless source
                    h2[a] = hn;
                }
            }
        }
    };

    // time loop unrolled by 2 so LDS buffer roles (and all addresses) are static
    for (int t = 0; t < TSTEPS; t += 2) {
        step(buf[0], buf[1], t);
        __syncthreads();
        if (wave == 0)
            out[(size_t)b * TSTEPS * NP + (size_t)t * NP + lane] =
                buf[1][pi * STRIDE + pj];

        step(buf[1], buf[0], t + 1);
        __syncthreads();
        if (wave == 0)
            out[(size_t)b * TSTEPS * NP + (size_t)(t + 1) * NP + lane] =
                buf[0][pi * STRIDE + pj];
    }
}

extern "C" void kernel_launch(void* const* d_in, const int* in_sizes, int n_in,
                              void* d_out, int out_size, void* d_ws, size_t ws_size,
                              hipStream_t stream) {
    const float* x       = (const float*)d_in[0];
    const float* cmat    = (const float*)d_in[1];
    const float* rho     = (const float*)d_in[2];
    const int*   src_i   = (const int*)d_in[3];
    const int*   src_j   = (const int*)d_in[4];
    const int*   probe_i = (const int*)d_in[5];
    const int*   probe_j = (const int*)d_in[6];
    float*       out     = (float*)d_out;

    wave_kernel<<<NB, 1024, 0, stream>>>(x, cmat, rho, src_i, src_j,
                                         probe_i, probe_j, out);
}